// AdaptiveMask_32487132627485
// MI455X (gfx1250) — compile-verified
//
#include <hip/hip_runtime.h>

#define AS1 __attribute__((address_space(1)))
#define AS3 __attribute__((address_space(3)))

typedef __attribute__((__vector_size__(4 * sizeof(int)))) int v4i;

#if defined(__has_builtin)
#if __has_builtin(__builtin_amdgcn_global_load_async_to_lds_b128)
#define HAVE_ASYNC_LDS 1
#endif
#if __has_builtin(__builtin_amdgcn_s_wait_asynccnt)
#define HAVE_WAIT_ASYNC 1
#endif
#endif

#ifndef HAVE_ASYNC_LDS
#define HAVE_ASYNC_LDS 0
#endif
#ifndef HAVE_WAIT_ASYNC
#define HAVE_WAIT_ASYNC 0
#endif

#if HAVE_WAIT_ASYNC
#define WAIT_ASYNC(n) __builtin_amdgcn_s_wait_asynccnt(n)
#else
#define WAIT_ASYNC(n) asm volatile("s_wait_asynccnt %0" ::"i"(n) : "memory")
#endif

constexpr int kThreads = 256;               // 8 wave32s per block
constexpr int kStageFloats = kThreads * 4;  // 1024 floats = 4 KB = 4 rows of 256
constexpr int kStagesPerBlock = 8;          // 32 KB contiguous per block

// out[b,c,i,j] = x * saturate((min(i,j,255-i,255-j) + 256*cv - 95) / 32).
// The (a+c)/32 transform is monotone, so it commutes with the min:
//   saturate((min(rowm,colm)+c)/32) = saturate(min((rowm+c)/32, (colm+c)/32))
// -> the 4 column terms are loop-invariant per thread; only the row term
//    changes per stage (row index advances by exactly 4).

__global__ void __launch_bounds__(kThreads)
AdaptiveMask_32487132627485_kernel(const float* __restrict__ x,
                                   const float* __restrict__ cvp,
                                   float* __restrict__ out) {
  __shared__ float lds[2][kStageFloats];
  const int t = threadIdx.x;
  const int lane4 = t * 4;                  // 0..1020, 16B aligned
  const float c = cvp[0] * 256.0f - 95.0f;  // uniform scalar load
  const float kInv32 = 0.03125f;
  const unsigned blockBase =
      blockIdx.x * (unsigned)(kStagesPerBlock * kStageFloats);  // < 2^24

  // Uniform per-block base pointers: every access below is
  //   uniform_ptr + (stage constant + lane4)
  // with signed (no-wrap) offsets, so each unrolled stage folds into the
  // same saddr/voffset pair plus a distinct immediate instruction offset.
  const float* __restrict__ xs = x + blockBase;
  float* __restrict__ os = out + blockBase;

  // Loop-invariant column masks (stage stride 1024 is a multiple of 256).
  const int j0 = lane4 & 255;
  const int a0 = min(j0 + 0, 255 - (j0 + 0));
  const int a1 = min(j0 + 1, 255 - (j0 + 1));
  const int a2 = min(j0 + 2, 255 - (j0 + 2));
  const int a3 = min(j0 + 3, 255 - (j0 + 3));
  const float cm0 = ((float)a0 + c) * kInv32;
  const float cm1 = ((float)a1 + c) * kInv32;
  const float cm2 = ((float)a2 + c) * kInv32;
  const float cm3 = ((float)a3 + c) * kInv32;

  // Row index at stage 0 for this thread; +4 per stage thereafter.
  const int irow0 = (int)((blockBase >> 8) & 255u) + (lane4 >> 8);

#if HAVE_ASYNC_LDS
  // Prologue: async-fetch stage 0 into LDS buffer 0 (per-lane 16B slot).
  __builtin_amdgcn_global_load_async_to_lds_b128(
      (AS1 v4i*)(xs + lane4), (AS3 v4i*)&lds[0][lane4], 0, 0);
#endif

#pragma unroll
  for (int s = 0; s < kStagesPerBlock; ++s) {
    const int off = s * kStageFloats + lane4;  // stage const + lane4 (signed)
    float4 v;
#if HAVE_ASYNC_LDS
    if (s + 1 < kStagesPerBlock) {
      // Kick off next stage, then wait only for the *current* stage's data
      // (async loads complete in order -> asynccnt <= 1 retires the older
      // load). Each thread reads only the 16B slot it wrote, so the double
      // buffer needs no barriers.
      __builtin_amdgcn_global_load_async_to_lds_b128(
          (AS1 v4i*)(xs + (off + kStageFloats)),
          (AS3 v4i*)&lds[(s + 1) & 1][lane4], 0, 0);
      WAIT_ASYNC(1);
    } else {
      WAIT_ASYNC(0);
    }
    asm volatile("" ::: "memory");  // don't hoist the LDS read above the wait
    v = *(const float4*)&lds[s & 1][lane4];  // ds_load_b128
#else
    __builtin_prefetch(xs + off + kStageFloats, 0, 0);  // global_prefetch_b8
    v = *(const float4*)(xs + off);                     // global_load_b128
#endif

    // Per-stage row term (only varying part of the mask).
    const int i = (irow0 + 4 * s) & 255;
    const float rm = ((float)min(i, 255 - i) + c) * kInv32;

    // saturate(min(rm, cm)) -> v_min_f32 with clamp; then one v_mul_f32.
    v.x *= __saturatef(fminf(rm, cm0));
    v.y *= __saturatef(fminf(rm, cm1));
    v.z *= __saturatef(fminf(rm, cm2));
    v.w *= __saturatef(fminf(rm, cm3));

    *(float4*)(os + off) = v;  // global_store_b128 (imm offset form)
  }
}

extern "C" void kernel_launch(void* const* d_in, const int* in_sizes, int n_in,
                              void* d_out, int out_size, void* d_ws,
                              size_t ws_size, hipStream_t stream) {
  const float* x = (const float*)d_in[0];
  const float* cv = (const float*)d_in[1];  // single-element array
  float* out = (float*)d_out;

  const int n = in_sizes[0];  // 32*8*256*256 = 16,777,216 (multiple of 8192)
  const int blocks = n / (kStageFloats * kStagesPerBlock);  // 2048

  AdaptiveMask_32487132627485_kernel<<<blocks, kThreads, 0, stream>>>(x, cv,
                                                                      out);
}